// InterAtomDistance_76768245449516
// MI455X (gfx1250) — compile-verified
//
#include <hip/hip_runtime.h>
#include <math.h>

typedef __attribute__((ext_vector_type(2))) float v2f;
typedef __attribute__((ext_vector_type(8))) float v8f;

// ---------------------------------------------------------------------------
// Kernel 1: one-shot pack of WMMA operand layouts + metadata.
//   rawP/rawT[g] = { x0, x1, x2, 1 }          (A-operand slots, K=3 pad = 1)
//   sclP/sclT[g] = { -2x0, -2x1, -2x2, |x|^2 } (B-operand slots, K=3 = sq)
//   meta4[g]     = { |p|^2, |t|^2, meta, 0 }
// meta = res_idx if kept else unique negative sentinel; pair validity
// (mask_i & mask_j & res_i==res_j) collapses to meta_i == meta_j (i>j
// excludes the self-pair). WMMA then yields G = -2*x.y + sq_j directly.
// Also zeroes the per-batch [sum, cnt] accumulators.
// ---------------------------------------------------------------------------
__global__ void iad_pack(const float* __restrict__ pred,
                         const float* __restrict__ target,
                         const int*   __restrict__ mask,
                         const int*   __restrict__ res_idx,
                         float4* __restrict__ rawP, float4* __restrict__ sclP,
                         float4* __restrict__ rawT, float4* __restrict__ sclT,
                         float4* __restrict__ meta4,
                         float*  __restrict__ acc,
                         int BL, int B) {
    int g = blockIdx.x * blockDim.x + threadIdx.x;
    if (g < BL) {
        float p0 = pred[3 * g + 0], p1 = pred[3 * g + 1], p2 = pred[3 * g + 2];
        float t0 = target[3 * g + 0], t1 = target[3 * g + 1], t2 = target[3 * g + 2];
        float sp = fmaf(p0, p0, fmaf(p1, p1, p2 * p2));
        float st = fmaf(t0, t0, fmaf(t1, t1, t2 * t2));
        rawP[g] = make_float4(p0, p1, p2, 1.0f);
        sclP[g] = make_float4(-2.0f * p0, -2.0f * p1, -2.0f * p2, sp);
        rawT[g] = make_float4(t0, t1, t2, 1.0f);
        sclT[g] = make_float4(-2.0f * t0, -2.0f * t1, -2.0f * t2, st);
        int m = (mask[g] != 0) ? res_idx[g] : -(g + 1);
        meta4[g] = make_float4(sp, st, __int_as_float(m), 0.0f);
    }
    if (g < 2 * B) acc[g] = 0.0f;
}

// ---------------------------------------------------------------------------
// Kernel 2: one wave32 per 16x16 tile of the strict lower triangle.
// V_WMMA_F32_16X16X4_F32 computes G = A[16x4] * B[4x16] = -2*x_i.x_j + sq_j.
// ---------------------------------------------------------------------------
__global__ __launch_bounds__(256)
void iad_tile_kernel(const float4* __restrict__ rawP,
                     const float4* __restrict__ sclP,
                     const float4* __restrict__ rawT,
                     const float4* __restrict__ sclT,
                     const float4* __restrict__ meta4,
                     float* __restrict__ acc,
                     int B, int L, int TT, float invTT) {
    const int gtid = blockIdx.x * blockDim.x + threadIdx.x;
    const int wave = gtid >> 5;          // wave32 on gfx1250
    const int lane = gtid & 31;
    if (wave >= B * TT) return;          // wave-uniform exit (EXEC stays full)

    // b = wave / TT via reciprocal multiply + fixup (avoids integer idiv)
    int b = (int)((float)wave * invTT);
    int t = wave - b * TT;
    if (t < 0)        { --b; t += TT; }
    else if (t >= TT) { ++b; t -= TT; }

    // Triangular decode t -> (ti, tj), ti >= tj (raw v_sqrt + fixups)
    int ti = (int)((__builtin_amdgcn_sqrtf(8.0f * (float)t + 1.0f) - 1.0f) * 0.5f);
    if ((ti + 1) * (ti + 2) / 2 <= t) ++ti;
    if (ti * (ti + 1) / 2 > t) --ti;
    const int tj = t - ti * (ti + 1) / 2;

    const bool hi   = (lane & 16) != 0;  // half-wave: K=0,1 vs K=2,3 slots
    const int  l16  = lane & 15;
    const int  base = b * L;
    const int  Lm1  = L - 1;

    // --- WMMA operands: 4 aligned B128 loads + selects ---
    const int rowA = min(ti * 16 + l16, Lm1);  // A rows    (tile-i atoms)
    const int rowB = min(tj * 16 + l16, Lm1);  // B columns (tile-j atoms)

    const float4 rA = rawP[base + rowA];
    const float4 rT = rawT[base + rowA];
    const float4 sB = sclP[base + rowB];
    const float4 sT4 = sclT[base + rowB];

    v2f aP, aT, bP, bT;
    aP[0] = hi ? rA.z : rA.x;   aP[1] = hi ? rA.w : rA.y;   // [x0,x1 | x2,1]
    aT[0] = hi ? rT.z : rT.x;   aT[1] = hi ? rT.w : rT.y;
    bP[0] = hi ? sB.z : sB.x;   bP[1] = hi ? sB.w : sB.y;   // [-2y | sq_j]
    bT[0] = hi ? sT4.z : sT4.x; bT[1] = hi ? sT4.w : sT4.y;

    // G[r] = -2 * x_i . x_j + sq_j   (fp32 WMMA, zero accumulator)
    v8f zero = {};
    v8f gP = __builtin_amdgcn_wmma_f32_16x16x4_f32(false, aP, false, bP,
                                                   (short)0, zero, false, false);
    v8f gT = __builtin_amdgcn_wmma_f32_16x16x4_f32(false, aT, false, bT,
                                                   (short)0, zero, false, false);

    // --- Column (j) metadata: only the meta word is needed now ---
    const int j  = tj * 16 + l16;
    const int mj = __float_as_int(meta4[base + min(j, Lm1)].z);

    // --- Epilogue: VGPR r holds M = r + 8*half, N = lane&15 ---
    const int rowbase = ti * 16 + (hi ? 8 : 0);
    const int diff    = rowbase - j;                 // i>j  <=>  diff > -r
    // Row metadata: one base address, 8 constant-offset B96 loads.
    const float4* rowMeta = meta4 + (base + min(rowbase, L - 8));

    float    sum  = 0.0f;
    unsigned long long nbits;
    int      cnt  = 0;
#pragma unroll
    for (int r = 0; r < 8; ++r) {
        const float4 rm = rowMeta[r];
        const bool valid = (diff > -r) && (rowbase + r < L) && (j < L) &&
                           (__float_as_int(rm.z) == mj);
        const float dp = __builtin_amdgcn_sqrtf(fmaxf(gP[r] + rm.x, 0.0f));
        const float dq = __builtin_amdgcn_sqrtf(fmaxf(gT[r] + rm.y, 0.0f));
        const float d  = valid ? (dq - dp) : 0.0f;
        sum = fmaf(d, d, sum);
        nbits = __ballot(valid);                     // wave-uniform mask
        cnt += (int)__popcll(nbits);
    }

    // --- Wave32 butterfly reduction for sum; cnt is already wave-uniform ---
#pragma unroll
    for (int off = 16; off > 0; off >>= 1) sum += __shfl_xor(sum, off, 32);

    if (lane == 0) {
        atomicAdd(&acc[2 * b + 0], sum);
        atomicAdd(&acc[2 * b + 1], (float)cnt);
    }
}

// ---------------------------------------------------------------------------
// Kernel 3: loss[b] = sqrt(sum / (n + 1e-6) + 1e-6)
// ---------------------------------------------------------------------------
__global__ void iad_finalize(const float* __restrict__ acc,
                             float* __restrict__ out, int B) {
    int b = blockIdx.x * blockDim.x + threadIdx.x;
    if (b < B) {
        out[b] = sqrtf(acc[2 * b] / (acc[2 * b + 1] + 1e-6f) + 1e-6f);
    }
}

// ---------------------------------------------------------------------------
// Launch
// ---------------------------------------------------------------------------
extern "C" void kernel_launch(void* const* d_in, const int* in_sizes, int n_in,
                              void* d_out, int out_size, void* d_ws, size_t ws_size,
                              hipStream_t stream) {
    const float* pred    = (const float*)d_in[0];
    const float* target  = (const float*)d_in[1];
    const int*   mask    = (const int*)d_in[2];
    const int*   res_idx = (const int*)d_in[3];
    float*       out     = (float*)d_out;

    const int B  = out_size;              // 4
    const int BL = in_sizes[2];           // B * L
    const int L  = BL / B;                // 2048
    const int T  = (L + 15) / 16;         // 128 tiles per dim
    const int TT = T * (T + 1) / 2;       // lower-triangle tiles per batch

    float4* rawP  = (float4*)d_ws;
    float4* sclP  = rawP + BL;
    float4* rawT  = sclP + BL;
    float4* sclT  = rawT + BL;
    float4* meta4 = sclT + BL;
    float*  acc   = (float*)(meta4 + BL);          // 2*B floats

    // 1) pack operands + metadata, zero accumulators
    {
        int threads = 256;
        int blocks  = (BL + threads - 1) / threads;
        iad_pack<<<blocks, threads, 0, stream>>>(pred, target, mask, res_idx,
                                                 rawP, sclP, rawT, sclT,
                                                 meta4, acc, BL, B);
    }
    // 2) WMMA tile pass: one wave per 16x16 triangle tile
    {
        long waves   = (long)B * TT;
        long threads = waves * 32;
        int  block   = 256;               // 8 waves per block
        int  blocks  = (int)((threads + block - 1) / block);
        iad_tile_kernel<<<blocks, block, 0, stream>>>(rawP, sclP, rawT, sclT,
                                                      meta4, acc, B, L, TT,
                                                      1.0f / (float)TT);
    }
    // 3) finalize
    {
        iad_finalize<<<1, 64, 0, stream>>>(acc, out, B);
    }
}